// FNO2D_70566312673752
// MI455X (gfx1250) — compile-verified
//
#include <hip/hip_runtime.h>
#include <math.h>

// ---------------- problem constants ----------------
#define HH 512
#define WW 512
#define HWTOT (HH * WW)          // 262144
#define IN_DIM 3
#define WIDTH 64
#define N_LAYERS 4
#define MM1 20
#define MM2 20
#define FC_DIM 128
#define NKX 40                   // 20 top + 20 bottom kx modes
#define KYP 24                   // ky pairs padded 20 -> 24 (so 48 cols, /16 aligned)
#define TCOLS 48                 // 2*KYP floats per complex row

typedef __attribute__((ext_vector_type(2))) float v2f;
typedef __attribute__((ext_vector_type(8))) float v8f;

// =====================================================================
// f32 GEMM on the WMMA pipe:  C(MxN) = A(MxK) * B(KxN) [+bias_m]
// [+addsrc] [tanh].  One wave32 per 64x16 C macro-tile: 4 accumulators
// share one B fragment per k-step (4x fewer strided B loads per WMMA).
// V_WMMA_F32_16X16X4_F32.  Requires M%64==0, N%16==0, K%4==0.
//
// Fragment layouts per CDNA5 ISA 7.12.2:
//   A 16x4 : lane l (row M=l&15) holds K = 2*(l>>4) + {vgpr0,vgpr1}
//   B 4x16 : vgpr j holds row K = j + 2*(l>>4), col N = l&15
//   C 16x16: vgpr g holds row M = g + 8*(l>>4), col N = l&15
// =====================================================================
__global__ __launch_bounds__(256)
void gemm_f32_wmma(const float* __restrict__ A, int lda,
                   const float* __restrict__ B, int ldb,
                   float* __restrict__ C, int ldc,
                   int M, int N, int K,
                   const float* __restrict__ bias,
                   const float* __restrict__ addsrc,
                   int act)
{
    const int wave = threadIdx.x >> 5;     // wave-uniform
    const int lane = threadIdx.x & 31;
    const int tilesN = N >> 4;
    const int tilesM = M >> 6;             // 64-row macro tiles
    const int tile = blockIdx.x * 8 + wave;
    if (tile >= tilesM * tilesN) return;   // wave-uniform exit: EXEC all-ones below

    const int tm = tile / tilesN;
    const int tn = tile - tm * tilesN;
    const int m0 = tm << 6;
    const int n0 = tn << 4;
    const int lrow = lane & 15;
    const int lhi  = lane >> 4;

    const float* A0   = A + (size_t)(m0 + lrow) * lda + 2 * lhi;
    const float* Bcol = B + (size_t)(2 * lhi) * ldb + n0 + lrow;
    const size_t aS = (size_t)16 * lda;    // 16-row stride between sub-tiles

    v8f acc0 = {}, acc1 = {}, acc2 = {}, acc3 = {};
    for (int k = 0; k < K; k += 4) {
        v2f b;
        b.x = Bcol[(size_t)k * ldb];                     // B[k+0+2*lhi][n]
        b.y = Bcol[(size_t)(k + 1) * ldb];               // B[k+1+2*lhi][n]
        v2f a0 = *(const v2f*)(A0 + k);
        v2f a1 = *(const v2f*)(A0 + aS + k);
        v2f a2 = *(const v2f*)(A0 + 2 * aS + k);
        v2f a3 = *(const v2f*)(A0 + 3 * aS + k);
        acc0 = __builtin_amdgcn_wmma_f32_16x16x4_f32(false, a0, false, b, (short)0, acc0, false, false);
        acc1 = __builtin_amdgcn_wmma_f32_16x16x4_f32(false, a1, false, b, (short)0, acc1, false, false);
        acc2 = __builtin_amdgcn_wmma_f32_16x16x4_f32(false, a2, false, b, (short)0, acc2, false, false);
        acc3 = __builtin_amdgcn_wmma_f32_16x16x4_f32(false, a3, false, b, (short)0, acc3, false, false);
    }

    v8f accs[4] = { acc0, acc1, acc2, acc3 };
#pragma unroll
    for (int s = 0; s < 4; ++s) {
#pragma unroll
        for (int g = 0; g < 8; ++g) {
            const int m = m0 + s * 16 + g + 8 * lhi;
            const size_t off = (size_t)m * ldc + n0 + lrow;
            float v = accs[s][g];
            if (bias)   v += bias[m];
            if (addsrc) v += addsrc[off];
            if (act)    v = tanhf(v);
            C[off] = v;
        }
    }
}

// ---------------- twiddle tables (phase reduced mod 512: exact) -------
__global__ void init_wf(float* __restrict__ Wf)   // (512 x 48): fwd rfft over W
{
    int idx = blockIdx.x * blockDim.x + threadIdx.x;
    if (idx >= WW * KYP) return;
    int w = idx / KYP, ky = idx - (idx / KYP) * KYP;
    float s = 0.f, c = 0.f;
    if (ky < MM2) {
        float ang = -6.28318530717958647692f * (float)((ky * w) & 511) * (1.0f / 512.0f);
        __sincosf(ang, &s, &c);      // e^{-i 2pi ky w/512}: re=c, im=s(= -sin)
    }
    Wf[w * TCOLS + 2 * ky]     = c;
    Wf[w * TCOLS + 2 * ky + 1] = s;
}

__global__ void init_binv(float* __restrict__ Bi) // (48 x 512): irfft over W + 1/(H*W)
{
    int idx = blockIdx.x * blockDim.x + threadIdx.x;
    if (idx >= KYP * WW) return;
    int ky = idx / WW, w = idx - ky * WW;
    float cr = 0.f, ci = 0.f;
    if (ky < MM2) {
        float scale = (ky == 0 ? 1.0f : 2.0f) / (float)(HH * WW);
        float ang = 6.28318530717958647692f * (float)((ky * w) & 511) * (1.0f / 512.0f);
        float s, c;
        __sincosf(ang, &s, &c);
        cr =  scale * c;             // Re coeff
        ci = -scale * s;             // Im coeff (Re(X e^{+i}) = Xr*cos - Xi*sin)
    }
    Bi[(2 * ky) * WW + w]     = cr;
    Bi[(2 * ky + 1) * WW + w] = ci;
}

// ---------------- fc0: (H,W,3) channel-last -> (64, HW) ---------------
__global__ void fc0_kernel(const float* __restrict__ x,
                           const float* __restrict__ w,
                           const float* __restrict__ b,
                           float* __restrict__ v)
{
    int idx = blockIdx.x * blockDim.x + threadIdx.x;
    if (idx >= WIDTH * HWTOT) return;
    int o = idx >> 18;               // / HW
    int p = idx & (HWTOT - 1);
    float s = b[o];
#pragma unroll
    for (int c = 0; c < IN_DIM; ++c)
        s += x[p * IN_DIM + c] * w[o * IN_DIM + c];
    v[idx] = s;
}

// ------- stage-2 forward: DFT over H for the 40 retained kx modes -----
// t[(c*512+h)*48 + 2ky..] -> xf[((c*40+kxi)*20+ky)*2..]
__global__ void stage2_fwd(const float* __restrict__ t, float* __restrict__ xf)
{
    int idx = blockIdx.x * blockDim.x + threadIdx.x;
    if (idx >= WIDTH * NKX * MM2) return;
    int c   = idx / (NKX * MM2);
    int r   = idx - c * (NKX * MM2);
    int kxi = r / MM2;
    int ky  = r - kxi * MM2;
    int kx  = (kxi < MM1) ? kxi : (472 + kxi);   // 492 + (kxi-20)
    const float* base = t + (size_t)(c * HH) * TCOLS + 2 * ky;
    float sr = 0.f, si = 0.f;
    for (int h = 0; h < HH; ++h) {
        float s, co;
        float ang = -6.28318530717958647692f * (float)((kx * h) & 511) * (1.0f / 512.0f);
        __sincosf(ang, &s, &co);
        float tr = base[h * TCOLS];
        float ti = base[h * TCOLS + 1];
        sr += tr * co - ti * s;
        si += tr * s + ti * co;
    }
    xf[2 * idx]     = sr;
    xf[2 * idx + 1] = si;
}

// ------- spectral 64x64 complex channel mix at each retained mode -----
__global__ void specmul(const float* __restrict__ xf,
                        const float* __restrict__ w1r, const float* __restrict__ w1i,
                        const float* __restrict__ w2r, const float* __restrict__ w2i,
                        float* __restrict__ yf)
{
    int idx = blockIdx.x * blockDim.x + threadIdx.x;
    if (idx >= WIDTH * NKX * MM2) return;
    int o   = idx / (NKX * MM2);
    int r   = idx - o * (NKX * MM2);
    int kxi = r / MM2;
    int ky  = r - kxi * MM2;
    const float* wr = (kxi < MM1) ? w1r : w2r;
    const float* wi = (kxi < MM1) ? w1i : w2i;
    int kx2 = (kxi < MM1) ? kxi : (kxi - MM1);
    size_t wbase = ((size_t)o * MM1 + kx2) * MM2 + ky;   // (c*64 + o)*400 + kx2*20 + ky
    float sr = 0.f, si = 0.f;
    for (int c = 0; c < WIDTH; ++c) {
        float xr = xf[((c * NKX + kxi) * MM2 + ky) * 2];
        float xi = xf[((c * NKX + kxi) * MM2 + ky) * 2 + 1];
        float wrv = wr[wbase + (size_t)c * (WIDTH * MM1 * MM2)];
        float wiv = wi[wbase + (size_t)c * (WIDTH * MM1 * MM2)];
        sr += xr * wrv - xi * wiv;
        si += xr * wiv + xi * wrv;
    }
    yf[2 * idx]     = sr;
    yf[2 * idx + 1] = si;
}

// ------- stage-1 inverse: inverse DFT over H (40 modes -> 512 rows) ---
// yf -> z[(o*512+h)*48 + 2ky..], pad cols zeroed
__global__ void stage1_inv(const float* __restrict__ yf, float* __restrict__ z)
{
    int idx = blockIdx.x * blockDim.x + threadIdx.x;
    if (idx >= WIDTH * HH * KYP) return;
    int o  = idx / (HH * KYP);
    int r  = idx - o * (HH * KYP);
    int h  = r / KYP;
    int ky = r - h * KYP;
    size_t row = (size_t)(o * HH + h) * TCOLS;
    if (ky >= MM2) {                       // zero the padding columns
        z[row + 2 * ky] = 0.f;
        z[row + 2 * ky + 1] = 0.f;
        return;
    }
    float sr = 0.f, si = 0.f;
    for (int kxi = 0; kxi < NKX; ++kxi) {
        int kx = (kxi < MM1) ? kxi : (472 + kxi);
        float s, co;
        float ang = 6.28318530717958647692f * (float)((kx * h) & 511) * (1.0f / 512.0f);
        __sincosf(ang, &s, &co);
        float yr = yf[((o * NKX + kxi) * MM2 + ky) * 2];
        float yi = yf[((o * NKX + kxi) * MM2 + ky) * 2 + 1];
        sr += yr * co - yi * s;
        si += yr * s + yi * co;
    }
    z[row + 2 * ky]     = sr;
    z[row + 2 * ky + 1] = si;
}

// ------- fused fc1(64->128)+tanh+fc2(128->1): avoids 134MB buffer -----
__global__ __launch_bounds__(256)
void fc12_kernel(const float* __restrict__ v,
                 const float* __restrict__ w1, const float* __restrict__ b1,
                 const float* __restrict__ w2, const float* __restrict__ b2,
                 float* __restrict__ out)
{
    __shared__ float w1s[FC_DIM * WIDTH];
    __shared__ float b1s[FC_DIM];
    __shared__ float w2s[FC_DIM];
    for (int i = threadIdx.x; i < FC_DIM * WIDTH; i += blockDim.x) w1s[i] = w1[i];
    for (int i = threadIdx.x; i < FC_DIM; i += blockDim.x) { b1s[i] = b1[i]; w2s[i] = w2[i]; }
    __syncthreads();
    int p = blockIdx.x * blockDim.x + threadIdx.x;
    if (p >= HWTOT) return;
    float vin[WIDTH];
#pragma unroll
    for (int c = 0; c < WIDTH; ++c) vin[c] = v[(size_t)c * HWTOT + p];
    float o2 = b2[0];
    for (int f = 0; f < FC_DIM; ++f) {
        float s = b1s[f];
#pragma unroll
        for (int c = 0; c < WIDTH; ++c) s += w1s[f * WIDTH + c] * vin[c];
        o2 += w2s[f] * tanhf(s);
    }
    out[p] = o2;
}

// =====================================================================
extern "C" void kernel_launch(void* const* d_in, const int* in_sizes, int n_in,
                              void* d_out, int out_size, void* d_ws, size_t ws_size,
                              hipStream_t stream)
{
    const float* x      = (const float*)d_in[0];
    const float* fc0_w  = (const float*)d_in[1];
    const float* fc0_b  = (const float*)d_in[2];
    const float* sw1_r  = (const float*)d_in[3];
    const float* sw1_i  = (const float*)d_in[4];
    const float* sw2_r  = (const float*)d_in[5];
    const float* sw2_i  = (const float*)d_in[6];
    const float* conv_w = (const float*)d_in[7];
    const float* conv_b = (const float*)d_in[8];
    const float* fc1_w  = (const float*)d_in[9];
    const float* fc1_b  = (const float*)d_in[10];
    const float* fc2_w  = (const float*)d_in[11];
    const float* fc2_b  = (const float*)d_in[12];
    float* out = (float*)d_out;

    // workspace carve-up (floats); total ~215 MB
    float* ws = (float*)d_ws;
    float* vA = ws;                                      // 64*HW
    float* vB = vA + (size_t)WIDTH * HWTOT;              // 64*HW
    float* x1 = vB + (size_t)WIDTH * HWTOT;              // 64*HW
    float* t  = x1 + (size_t)WIDTH * HWTOT;              // 32768*48
    float* z  = t  + (size_t)(WIDTH * HH) * TCOLS;       // 32768*48
    float* xf = z  + (size_t)(WIDTH * HH) * TCOLS;       // 64*800*2
    float* yf = xf + (size_t)WIDTH * NKX * MM2 * 2;      // 64*800*2
    float* Wf = yf + (size_t)WIDTH * NKX * MM2 * 2;      // 512*48
    float* Bi = Wf + (size_t)WW * TCOLS;                 // 48*512

    const size_t lw_off = (size_t)WIDTH * WIDTH * MM1 * MM2; // per-layer spectral wt stride

    init_wf  <<<(WW * KYP + 255) / 256, 256, 0, stream>>>(Wf);
    init_binv<<<(KYP * WW + 255) / 256, 256, 0, stream>>>(Bi);

    fc0_kernel<<<(WIDTH * HWTOT + 255) / 256, 256, 0, stream>>>(x, fc0_w, fc0_b, vA);

    for (int l = 0; l < N_LAYERS; ++l) {
        // (1) truncated rfft over W as GEMM: t(32768x48) = vA(32768x512) * Wf(512x48)
        int tiles = (WIDTH * HH / 64) * (TCOLS / 16);
        gemm_f32_wmma<<<(tiles + 7) / 8, 256, 0, stream>>>(
            vA, WW, Wf, TCOLS, t, TCOLS,
            WIDTH * HH, TCOLS, WW, nullptr, nullptr, 0);

        // (2) DFT over H at the 40 retained kx modes
        stage2_fwd<<<(WIDTH * NKX * MM2 + 255) / 256, 256, 0, stream>>>(t, xf);

        // (3) complex 64x64 channel mix per mode
        specmul<<<(WIDTH * NKX * MM2 + 255) / 256, 256, 0, stream>>>(
            xf, sw1_r + l * lw_off, sw1_i + l * lw_off,
                sw2_r + l * lw_off, sw2_i + l * lw_off, yf);

        // (4) inverse DFT over H (40 modes -> 512)
        stage1_inv<<<(WIDTH * HH * KYP + 255) / 256, 256, 0, stream>>>(yf, z);

        // (5) irfft over W as GEMM: x1(32768x512) = z(32768x48) * Bi(48x512)
        tiles = (WIDTH * HH / 64) * (WW / 16);
        gemm_f32_wmma<<<(tiles + 7) / 8, 256, 0, stream>>>(
            z, TCOLS, Bi, WW, x1, WW,
            WIDTH * HH, WW, TCOLS, nullptr, nullptr, 0);

        // (6) pointwise conv as GEMM + fused (x1 + bias, tanh) epilogue:
        //     vB(64xHW) = conv_w(64x64) * vA(64xHW) + x1 + b, tanh
        //     M=64 -> each wave covers the full M extent (one B stream)
        tiles = (WIDTH / 64) * (HWTOT / 16);
        gemm_f32_wmma<<<(tiles + 7) / 8, 256, 0, stream>>>(
            conv_w + (size_t)l * WIDTH * WIDTH, WIDTH,
            vA, HWTOT, vB, HWTOT,
            WIDTH, HWTOT, WIDTH,
            conv_b + (size_t)l * WIDTH, x1, 1);

        float* tmp = vA; vA = vB; vB = tmp;
    }

    // fc1 + tanh + fc2 fused, one pass over the activation tensor
    fc12_kernel<<<(HWTOT + 255) / 256, 256, 0, stream>>>(
        vA, fc1_w, fc1_b, fc2_w, fc2_b, out);
}